// BiLSTM_CRF_53171695124874
// MI455X (gfx1250) — compile-verified
//
#include <hip/hip_runtime.h>
#include <hip/hip_bf16.h>

typedef __attribute__((ext_vector_type(2))) float v2f;
typedef __attribute__((ext_vector_type(8))) float v8f;

#define CRF_B     64
#define CRF_S     512
#define CRF_H     768
#define CRF_T     11
#define CRF_TP    16      // padded tag dim (one WMMA N tile)
#define CRF_START 9
#define CRF_STOP  10
#define CRF_NEG   (-10000.0f)
#define FNEG_INF  (-3.0e38f)

#define LDSW_STRIDE 772   // 768 + 4 pad: lane l bank = (l*772)%64 = 4l -> conflict free

// ---------------------------------------------------------------------------
// Kernel 1: feats[b,s,t] = hs[b,s,:] @ W[:,t] + b[t], t padded to 16 (pad = 0)
// One wave per 16-row M tile, fp32 WMMA 16x16x4, K loop over 768 (192 WMMAs).
// ---------------------------------------------------------------------------
__global__ __launch_bounds__(256) void feats_gemm_wmma(
    const float* __restrict__ hs, const float* __restrict__ Wm,
    const float* __restrict__ bias, float* __restrict__ feats)
{
  __shared__ float sW[CRF_TP * LDSW_STRIDE];   // W transposed: sW[n][h]
  const int tid = threadIdx.x;

  for (int idx = tid; idx < CRF_TP * CRF_H; idx += 256) {
    int n = idx / CRF_H;
    int h = idx - n * CRF_H;
    sW[n * LDSW_STRIDE + h] = (n < CRF_T) ? Wm[h * CRF_T + n] : 0.0f;
  }
  __syncthreads();

  const int wave = tid >> 5;
  const int lane = tid & 31;
  const int hb   = lane >> 4;   // half-wave selects K subgroup (K+2*hb)
  const int l    = lane & 15;   // M row (A) / N col (B,C,D) within tile

  const long tile = (long)blockIdx.x * 8 + wave;       // 2048 tiles total
  const long row  = tile * 16 + l;

  const float* arow = hs + row * CRF_H + 2 * hb;       // A frag: M=l, K=k+2*hb..+1
  const float* brow = sW + l * LDSW_STRIDE + 2 * hb;   // B frag: N=l, K=k+2*hb..+1

  v8f acc;
  const float bv = (l < CRF_T) ? bias[l] : 0.0f;       // fold bias into C
  #pragma unroll
  for (int r = 0; r < 8; ++r) acc[r] = bv;

  // prefetch this wave's A rows (emits global_prefetch_b8)
  for (int k = 0; k < CRF_H; k += 64) __builtin_prefetch(arow + k, 0, 0);

  #pragma unroll 4
  for (int k = 0; k < CRF_H; k += 4) {
    v2f a  = *(const v2f*)(arow + k);
    v2f bf = *(const v2f*)(brow + k);
    // 8 args: (neg_a, A, neg_b, B, c_mod, C, reuse_a, reuse_b)
    acc = __builtin_amdgcn_wmma_f32_16x16x4_f32(false, a, false, bf,
                                                (short)0, acc, false, false);
  }

  // D layout: reg r, lane -> M = r + 8*hb, N = l
  float* ob = feats + (tile * 16 + 8 * hb) * CRF_TP + l;
  #pragma unroll
  for (int r = 0; r < 8; ++r) ob[r * CRF_TP] = acc[r];
}

// ---------------------------------------------------------------------------
// Kernel 2: CRF scans. grid (B, 2): y==0 forward+gold (writes diffs[b]),
// y==1 Viterbi+backtrace (writes path_score and best_path). One wave32 per
// block; state fv[j] lives in lane j, broadcast via __shfl.
// ---------------------------------------------------------------------------
__global__ __launch_bounds__(32) void crf_scan_kernel(
    const float* __restrict__ feats, const int* __restrict__ amask,
    const int* __restrict__ tags, const float* __restrict__ trans,
    float* __restrict__ out, float* __restrict__ diffs)
{
  const int b    = blockIdx.x;
  const int mode = blockIdx.y;     // 0 = forward/gold, 1 = viterbi
  const int j    = threadIdx.x;    // state index (valid j < 11)

  __shared__ float sTrans[CRF_T * CRF_T];
  __shared__ unsigned char sBptr[CRF_S * CRF_TP];

  for (int i = j; i < CRF_T * CRF_T; i += 32) sTrans[i] = trans[i];
  __syncthreads();

  // tcol[i] = trans[i][j] (forward), trow[i] = trans[j][i] (viterbi)
  float tcol[CRF_T], trow[CRF_T];
  if (j < CRF_T) {
    #pragma unroll
    for (int i = 0; i < CRF_T; ++i) {
      tcol[i] = sTrans[i * CRF_T + j];
      trow[i] = sTrans[j * CRF_T + i];
    }
  } else {
    #pragma unroll
    for (int i = 0; i < CRF_T; ++i) { tcol[i] = CRF_NEG; trow[i] = CRF_NEG; }
  }

  const float* fb = feats + (long)b * CRF_S * CRF_TP;
  float fv = (j == CRF_START) ? 0.0f : CRF_NEG;

  if (mode == 0) {
    // ---------------- forward (logsumexp) scan ----------------
    for (int s = 0; s < CRF_S; ++s) {
      const float feat = (j < CRF_TP) ? fb[s * CRF_TP + j] : 0.0f;
      float fm = FNEG_INF;
      float sc[CRF_T];
      #pragma unroll
      for (int i = 0; i < CRF_T; ++i) {
        float fi = __shfl(fv, i, 32);
        float v  = fi + tcol[i];
        sc[i] = v;
        fm = fmaxf(fm, v);
      }
      float fsum = 0.0f;
      #pragma unroll
      for (int i = 0; i < CRF_T; ++i) fsum += __expf(sc[i] - fm);
      const float fnew = fm + __logf(fsum) + feat;
      if (amask[b * CRF_S + s] == 1) fv = fnew;
    }
    // forward_score = logsumexp_j( fv[j] + trans[STOP][j] )
    float term = (j < CRF_T) ? fv + sTrans[CRF_STOP * CRF_T + j] : FNEG_INF;
    float tmax = term;
    for (int off = 16; off; off >>= 1) tmax = fmaxf(tmax, __shfl_xor(tmax, off, 32));
    float e = (j < CRF_T) ? __expf(term - tmax) : 0.0f;
    for (int off = 16; off; off >>= 1) e += __shfl_xor(e, off, 32);
    const float forward_score = tmax + __logf(e);

    // gold score, parallel over s across lanes
    float g = 0.0f;
    for (int s = j; s < CRF_S; s += 32) {
      const int cur  = tags[b * CRF_S + s];
      const int prev = (s == 0) ? CRF_START : tags[b * CRF_S + s - 1];
      if (amask[b * CRF_S + s] == 1)
        g += sTrans[cur * CRF_T + prev] + fb[s * CRF_TP + cur];
    }
    for (int off = 16; off; off >>= 1) g += __shfl_xor(g, off, 32);
    if (j == 0) {
      g += sTrans[CRF_STOP * CRF_T + tags[b * CRF_S + (CRF_S - 1)]];
      diffs[b] = forward_score - g;
    }
  } else {
    // ---------------- Viterbi scan (reference ignores mask here) ----------
    for (int s = 0; s < CRF_S; ++s) {
      const float feat = (j < CRF_TP) ? fb[s * CRF_TP + j] : 0.0f;
      float vm = FNEG_INF;
      int varg = 0;
      #pragma unroll
      for (int i = 0; i < CRF_T; ++i) {
        float vi = __shfl(fv, i, 32);
        float v  = vi + trow[i];           // trans[j][i], as in reference vstep
        if (v > vm) { vm = v; varg = i; }
      }
      fv = vm + feat;
      if (j < CRF_TP) sBptr[s * CRF_TP + j] = (unsigned char)varg;
    }
    // terminal: path_score = max_j(fv[j] + trans[STOP][j]), best_last = argmax
    float vterm = (j < CRF_T) ? fv + sTrans[CRF_STOP * CRF_T + j] : FNEG_INF;
    float pmax = vterm;
    for (int off = 16; off; off >>= 1) pmax = fmaxf(pmax, __shfl_xor(pmax, off, 32));
    int cand = (j < CRF_T && vterm == pmax) ? j : 1000;
    for (int off = 16; off; off >>= 1) cand = min(cand, __shfl_xor(cand, off, 32));

    __syncthreads();
    if (j == 0) {
      out[1 + b] = pmax;
      int tag = cand;
      float* bp = out + 1 + CRF_B + (long)b * CRF_S;
      bp[CRF_S - 1] = (float)tag;
      for (int s = CRF_S - 1; s >= 1; --s) {
        tag = sBptr[s * CRF_TP + tag];
        bp[s - 1] = (float)tag;
      }
    }
  }
}

// ---------------------------------------------------------------------------
// Kernel 3: nll = mean over batch of diffs
// ---------------------------------------------------------------------------
__global__ __launch_bounds__(32) void nll_reduce_kernel(
    const float* __restrict__ diffs, float* __restrict__ out)
{
  const int lane = threadIdx.x;
  float v = diffs[lane] + diffs[lane + 32];
  for (int off = 16; off; off >>= 1) v += __shfl_xor(v, off, 32);
  if (lane == 0) out[0] = v * (1.0f / (float)CRF_B);
}

extern "C" void kernel_launch(void* const* d_in, const int* in_sizes, int n_in,
                              void* d_out, int out_size, void* d_ws, size_t ws_size,
                              hipStream_t stream)
{
  const float* hs    = (const float*)d_in[0];   // (B,S,H) f32
  const int*   amask = (const int*)d_in[1];     // (B,S) i32
  const int*   tags  = (const int*)d_in[2];     // (B,S) i32
  const float* Wm    = (const float*)d_in[3];   // (H,T) f32
  const float* bias  = (const float*)d_in[4];   // (T,) f32
  const float* trans = (const float*)d_in[5];   // (T,T) f32

  float* out   = (float*)d_out;                 // [nll | path_score(64) | best_path(64*512)]
  float* feats = (float*)d_ws;                                          // 32768*16 f32 = 2 MB
  float* diffs = (float*)((char*)d_ws + (size_t)CRF_B * CRF_S * CRF_TP * sizeof(float));

  // 2048 M-tiles / 8 waves per block = 256 blocks
  feats_gemm_wmma<<<dim3(256), dim3(256), 0, stream>>>(hs, Wm, bias, feats);
  crf_scan_kernel<<<dim3(CRF_B, 2), dim3(32), 0, stream>>>(feats, amask, tags, trans, out, diffs);
  nll_reduce_kernel<<<dim3(1), dim3(32), 0, stream>>>(diffs, out);
}